// Correlation_7404523619037
// MI455X (gfx1250) — compile-verified
//
#include <hip/hip_runtime.h>

// FlowNetC correlation as banded GEMM on CDNA5 WMMA (f32 16x16x4).
// out[b, dy*21+dx, y, x] = (1/C) * sum_c in1[b,c,y,x] * in2[b,c,y+oy,x+ox]
// Band of Gram matrix G[x,u] = in1_row^T @ in2_row, |u-x| <= 20.

typedef float v2f __attribute__((ext_vector_type(2)));
typedef float v8f __attribute__((ext_vector_type(8)));

#define CH    256
#define HH    96
#define WW    128
#define RAD   10
#define DISP  21          // 2*RAD+1
#define KC    32          // K chunk staged in LDS
#define NCHUNK (CH / KC)  // 8
#define LSTR  36          // LDS row stride (floats) for sA/sB: 36m mod 64 hits all banks
#define GSTR  84          // LDS row stride (floats) for G band staging (80 cols + pad)
#define CHW   (HH * WW)

__global__ __launch_bounds__(256)
void corr_wmma_kernel(const float* __restrict__ in1,
                      const float* __restrict__ in2,
                      float* __restrict__ out)
{
    // 43008 B static LDS. During K loop: sA = [0,4608), sB = [4608,9216).
    // After K loop (barrier-separated): whole buffer reused as G band [128][84].
    __shared__ float smem[WW * GSTR];
    float* sA = smem;
    float* sB = smem + WW * LSTR;

    const int dyi = blockIdx.x;   // 0..20
    const int y   = blockIdx.y;   // 0..95
    const int b   = blockIdx.z;   // 0..B-1
    const int tid = threadIdx.x;

    const int oy = (dyi - RAD) * 2;
    const int y2 = y + oy;

    // out base at channel (dyi*21 + 0), row y
    const long outBase = (((long)b * (DISP * DISP) + (long)dyi * DISP) * HH + y) * WW;

    if (y2 < 0 || y2 >= HH) {
        // zero-padded in2 rows: whole 21 x 128 output slab is zero
        for (int idx = tid; idx < DISP * WW; idx += 256) {
            const int dx = idx >> 7;
            const int x  = idx & (WW - 1);
            out[outBase + (long)dx * CHW + x] = 0.0f;
        }
        return;
    }

    const int lane = tid & 31;
    const int w    = tid >> 5;    // wave id == x-tile index (8 tiles over W=128)
    const int m    = lane & 15;
    const int half = lane >> 4;   // 0: K=0,1  1: K=2,3 (32-bit A/B fragment layout)
    const int x0   = w * 16;

    // A fragment base: A[i,k] = in1[c0+k, x0+i], lane m -> row, half -> K pair
    const int aBase = (x0 + m) * LSTR + 2 * half;

    // 5 u-tiles per x-tile (band |u-x| <= 20 spans U in [X-2, X+2]).
    // Clamp out-of-range tiles to a valid address; their results are never read.
    int bBase[5];
#pragma unroll
    for (int uu = 0; uu < 5; ++uu) {
        int U = w - 2 + uu;
        U = U < 0 ? 0 : (U > 7 ? 7 : U);
        bBase[uu] = (U * 16 + m) * LSTR + 2 * half;
    }

    v8f acc[5] = {};

    // staging-load thread mapping: lane-contiguous x for coalesced global loads
    const int lx  = tid & (WW - 1);   // 0..127
    const int lc0 = tid >> 7;         // 0..1

    const float* g1 = in1 + (((long)b * CH) * HH + y ) * WW;  // + c*H*W + x
    const float* g2 = in2 + (((long)b * CH) * HH + y2) * WW;

    for (int kc = 0; kc < NCHUNK; ++kc) {
        __syncthreads();  // previous chunk fully consumed
        const int cb = kc * KC;
#pragma unroll
        for (int cc = lc0; cc < KC; cc += 2) {
            // transpose into LDS: sA[x][c]; stores are bank-conflict-free
            sA[lx * LSTR + cc] = g1[(long)(cb + cc) * CHW + lx];
            sB[lx * LSTR + cc] = g2[(long)(cb + cc) * CHW + lx];
        }
        if (kc + 1 < NCHUNK) {
            __builtin_prefetch(&g1[(long)(cb + KC + lc0) * CHW + lx], 0, 1);
            __builtin_prefetch(&g2[(long)(cb + KC + lc0) * CHW + lx], 0, 1);
        }
        __syncthreads();

#pragma unroll
        for (int kk = 0; kk < KC; kk += 4) {
            v2f a;
            a.x = sA[aBase + kk];
            a.y = sA[aBase + kk + 1];
#pragma unroll
            for (int uu = 0; uu < 5; ++uu) {
                v2f bb;
                bb.x = sB[bBase[uu] + kk];
                bb.y = sB[bBase[uu] + kk + 1];
                // D = A(16x4 f32) * B(4x16 f32) + C(16x16 f32)
                acc[uu] = __builtin_amdgcn_wmma_f32_16x16x4_f32(
                    false, a, false, bb, (short)0, acc[uu], false, false);
            }
        }
    }

    __syncthreads();  // all waves done reading sA/sB
    // Stage G band: G[x][j], j = (U - X + 2)*16 + (u & 15).
    // C/D layout: VGPR r -> row (r + 8*half), lane m -> col.
#pragma unroll
    for (int uu = 0; uu < 5; ++uu) {
#pragma unroll
        for (int r = 0; r < 8; ++r) {
            smem[(x0 + r + 8 * half) * GSTR + uu * 16 + m] = acc[uu][r];
        }
    }
    __syncthreads();

    // Extract 21 diagonals; coalesced 128-wide writes per dx.
    const float scale = 1.0f / (float)CH;
    for (int idx = tid; idx < DISP * WW; idx += 256) {
        const int dx = idx >> 7;
        const int x  = idx & (WW - 1);
        const int u  = x + (dx - RAD) * 2;
        float v = 0.0f;
        if (u >= 0 && u < WW) {
            const int X = x >> 4;
            const int U = u >> 4;          // always in [X-2, X+2]
            v = smem[x * GSTR + (U - X + 2) * 16 + (u & 15)] * scale;
        }
        out[outBase + (long)dx * CHW + x] = v;
    }
}

extern "C" void kernel_launch(void* const* d_in, const int* in_sizes, int n_in,
                              void* d_out, int out_size, void* d_ws, size_t ws_size,
                              hipStream_t stream) {
    const float* in1 = (const float*)d_in[0];
    const float* in2 = (const float*)d_in[1];
    float* out = (float*)d_out;

    const int B = in_sizes[0] / (CH * HH * WW);   // 8 for the reference shapes
    dim3 grid(DISP, HH, B);
    dim3 block(256);
    corr_wmma_kernel<<<grid, block, 0, stream>>>(in1, in2, out);
}